// GCN_87308095193263
// MI455X (gfx1250) — compile-verified
//
#include <hip/hip_runtime.h>
#include <hip/hip_bf16.h>
#include <math.h>

typedef __attribute__((ext_vector_type(2))) float v2f;
typedef __attribute__((ext_vector_type(8))) float v8f;

// ---------------------------------------------------------------------------
// Degree / normalization: deg[i] = 1 (self loop) + #incoming edges; dinv = rsqrt
// ---------------------------------------------------------------------------
__global__ void deg_init_kernel(float* __restrict__ deg, int n) {
  int i = blockIdx.x * blockDim.x + threadIdx.x;
  if (i < n) deg[i] = 1.0f;  // self-loop contribution
}

__global__ void deg_accum_kernel(const long long* __restrict__ dst,
                                 float* __restrict__ deg, int e) {
  int i = blockIdx.x * blockDim.x + threadIdx.x;
  int stride = gridDim.x * blockDim.x;
  for (; i < e; i += stride) {
    unsafeAtomicAdd(&deg[(int)dst[i]], 1.0f);  // hw global_atomic_add_f32
  }
}

__global__ void dinv_kernel(float* __restrict__ deg, int n) {
  int i = blockIdx.x * blockDim.x + threadIdx.x;
  if (i < n) deg[i] = rsqrtf(deg[i]);  // deg >= 1 always (self loops)
}

// ---------------------------------------------------------------------------
// Dense H = act(X) @ W via V_WMMA_F32_16X16X4_F32.
// One wave per 16-row tile; loops over 16-col tiles of W; K unrolled in 4s.
// A 16x4 f32 layout: lane l(<16),half h: v0=X[m0+l][kb+2h], v1=X[m0+l][kb+2h+1]
// B 4x16  f32 layout: v0=W[kb+2h][f0+l],  v1=W[kb+2h+1][f0+l]
// D 16x16: VGPR r -> row m0+r+8h, col f0+l
// ---------------------------------------------------------------------------
template <int K, int F, bool RELU>
__global__ void gemm_wmma_kernel(const float* __restrict__ X,
                                 const float* __restrict__ W,
                                 float* __restrict__ H, int n) {
  const int wave   = blockIdx.x * (blockDim.x >> 5) + (threadIdx.x >> 5);
  const int ntiles = n >> 4;
  if (wave >= ntiles) return;  // wave-uniform: EXEC stays all-ones for WMMA

  const int lane = threadIdx.x & 31;
  const int half = lane >> 4;
  const int l    = lane & 15;
  const int m0   = wave << 4;

#pragma unroll
  for (int f0 = 0; f0 < F; f0 += 16) {
    v8f c = {};
#pragma unroll
    for (int kb = 0; kb < K; kb += 4) {
      const int ka = kb + 2 * half;
      float a0 = X[(m0 + l) * K + ka];
      float a1 = X[(m0 + l) * K + ka + 1];
      if (RELU) { a0 = fmaxf(a0, 0.0f); a1 = fmaxf(a1, 0.0f); }
      v2f a; a.x = a0; a.y = a1;
      v2f b;
      b.x = W[ka * F + f0 + l];
      b.y = W[(ka + 1) * F + f0 + l];
      c = __builtin_amdgcn_wmma_f32_16x16x4_f32(false, a, false, b,
                                                (short)0, c, false, false);
    }
#pragma unroll
    for (int r = 0; r < 8; ++r) {
      H[(m0 + r + 8 * half) * F + f0 + l] = c[r];
    }
  }
}

// ---------------------------------------------------------------------------
// Aggregation init: A[i,f] = H[i,f]*dinv[i]^2 (self loop) + bias[f]
// ---------------------------------------------------------------------------
__global__ void agg_init_kernel(const float* __restrict__ H,
                                const float* __restrict__ dinv,
                                const float* __restrict__ bias,
                                float* __restrict__ A, int n, int F) {
  int idx = blockIdx.x * blockDim.x + threadIdx.x;
  int total = n * F;
  int stride = gridDim.x * blockDim.x;
  for (; idx < total; idx += stride) {
    int i = idx / F;
    int f = idx - i * F;
    float di = dinv[i];
    A[idx] = H[idx] * di * di + bias[f];
  }
}

// ---------------------------------------------------------------------------
// Edge scatter: one wave per edge, lanes over features (coalesced row gather)
// A[dst,f] += H[src,f] * dinv[src]*dinv[dst]
// Edge indices are wave-uniform -> readfirstlane into SGPRs (scalar addressing)
// ---------------------------------------------------------------------------
template <int F>
__global__ void agg_edge_kernel(const float* __restrict__ H,
                                const float* __restrict__ dinv,
                                const long long* __restrict__ src,
                                const long long* __restrict__ dst,
                                float* __restrict__ A, int e) {
  int wave   = blockIdx.x * (blockDim.x >> 5) + (threadIdx.x >> 5);
  int nwaves = gridDim.x * (blockDim.x >> 5);
  int lane   = threadIdx.x & 31;
  for (int ed = wave; ed < e; ed += nwaves) {
    int s = __builtin_amdgcn_readfirstlane((int)src[ed]);
    int d = __builtin_amdgcn_readfirstlane((int)dst[ed]);
    float norm = dinv[s] * dinv[d];
#pragma unroll
    for (int f = lane; f < F; f += 32) {
      unsafeAtomicAdd(&A[d * F + f], H[s * F + f] * norm);
    }
  }
}

// ---------------------------------------------------------------------------
// Layer 3: h3[i] = relu(A2[i,:]) . W3   (F_out = 1; scalar dot, WMMA wasteful)
// ---------------------------------------------------------------------------
__global__ void layer3_dot_kernel(const float* __restrict__ A2,
                                  const float* __restrict__ W3,
                                  float* __restrict__ H3, int n) {
  int i = blockIdx.x * blockDim.x + threadIdx.x;
  if (i >= n) return;
  float sum = 0.0f;
#pragma unroll
  for (int j = 0; j < 64; ++j) {
    sum += fmaxf(A2[i * 64 + j], 0.0f) * W3[j];
  }
  H3[i] = sum;
}

__global__ void agg3_init_kernel(const float* __restrict__ H3,
                                 const float* __restrict__ dinv,
                                 const float* __restrict__ b3,
                                 float* __restrict__ O, int n) {
  int i = blockIdx.x * blockDim.x + threadIdx.x;
  if (i < n) {
    float di = dinv[i];
    O[i] = H3[i] * di * di + b3[0];
  }
}

__global__ void agg3_edge_kernel(const float* __restrict__ H3,
                                 const float* __restrict__ dinv,
                                 const long long* __restrict__ src,
                                 const long long* __restrict__ dst,
                                 float* __restrict__ O, int e) {
  int i = blockIdx.x * blockDim.x + threadIdx.x;
  int stride = gridDim.x * blockDim.x;
  for (; i < e; i += stride) {
    int s = (int)src[i];
    int d = (int)dst[i];
    unsafeAtomicAdd(&O[d], H3[s] * dinv[s] * dinv[d]);
  }
}

__global__ void sigmoid_kernel(const float* __restrict__ O,
                               float* __restrict__ out, int n) {
  int i = blockIdx.x * blockDim.x + threadIdx.x;
  if (i < n) out[i] = 1.0f / (1.0f + expf(-O[i]));
}

// ---------------------------------------------------------------------------
extern "C" void kernel_launch(void* const* d_in, const int* in_sizes, int n_in,
                              void* d_out, int out_size, void* d_ws, size_t ws_size,
                              hipStream_t stream) {
  const float*     x  = (const float*)d_in[0];
  const long long* ei = (const long long*)d_in[1];  // int64 [2, E]
  const float*     W1 = (const float*)d_in[2];
  const float*     b1 = (const float*)d_in[3];
  const float*     W2 = (const float*)d_in[4];
  const float*     b2 = (const float*)d_in[5];
  const float*     W3 = (const float*)d_in[6];
  const float*     b3 = (const float*)d_in[7];
  float*           out = (float*)d_out;

  const int n = in_sizes[0] / 16;   // 100000
  const int e = in_sizes[1] / 2;    // 3200000
  const long long* src = ei;
  const long long* dst = ei + e;

  // Workspace layout (bytes): dinv[n] | bufA[n*64] | bufB[n*64]
  char* ws = (char*)d_ws;
  size_t off0 = 0;
  size_t offA = (off0 + (size_t)n * 4 + 255) & ~(size_t)255;
  size_t offB = (offA + (size_t)n * 64 * 4 + 255) & ~(size_t)255;
  float* dinv = (float*)(ws + off0);
  float* bufA = (float*)(ws + offA);
  float* bufB = (float*)(ws + offB);

  const int T = 256;
  const int gn = (n + T - 1) / T;

  // --- normalization (shared by all layers) ---
  deg_init_kernel<<<gn, T, 0, stream>>>(dinv, n);
  deg_accum_kernel<<<2048, T, 0, stream>>>(dst, dinv, e);
  dinv_kernel<<<gn, T, 0, stream>>>(dinv, n);

  // --- layer 1: 16 -> 32 ---
  {
    int ntiles = n / 16;                       // 6250 (exact)
    int blocks = (ntiles + 7) / 8;             // 8 waves / 256-thread block
    gemm_wmma_kernel<16, 32, false><<<blocks, T, 0, stream>>>(x, W1, bufA, n);
    agg_init_kernel<<<4096, T, 0, stream>>>(bufA, dinv, b1, bufB, n, 32);
    agg_edge_kernel<32><<<4096, T, 0, stream>>>(bufA, dinv, src, dst, bufB, e);
  }

  // --- layer 2: 32 -> 64 (ReLU fused into A-load) ---
  {
    int ntiles = n / 16;
    int blocks = (ntiles + 7) / 8;
    gemm_wmma_kernel<32, 64, true><<<blocks, T, 0, stream>>>(bufB, W2, bufA, n);
    agg_init_kernel<<<4096, T, 0, stream>>>(bufA, dinv, b2, bufB, n, 64);
    agg_edge_kernel<64><<<4096, T, 0, stream>>>(bufA, dinv, src, dst, bufB, e);
  }

  // --- layer 3: 64 -> 1 (scalar dot) + sigmoid ---
  {
    float* h3 = bufA;            // reuse: only n floats needed
    float* o3 = bufA + n;        // aggregation target
    layer3_dot_kernel<<<gn, T, 0, stream>>>(bufB, W3, h3, n);
    agg3_init_kernel<<<gn, T, 0, stream>>>(h3, dinv, b3, o3, n);
    agg3_edge_kernel<<<4096, T, 0, stream>>>(h3, dinv, src, dst, o3, e);
    sigmoid_kernel<<<gn, T, 0, stream>>>(o3, out, n);
  }
}